// BiMap_11605001634476
// MI455X (gfx1250) — compile-verified
//
#include <hip/hip_runtime.h>
#include <math.h>

// Problem dims (match reference)
#define B_  32
#define C_  16
#define NI  64
#define NO  32
#define M_  4

typedef float v2f __attribute__((ext_vector_type(2)));
typedef float v8f __attribute__((ext_vector_type(8)));
typedef int   v4i __attribute__((ext_vector_type(4)));

// ---------------------------------------------------------------------------
// Optional CDNA5 async global->LDS direct copy (ASYNCcnt path, bypasses VGPRs).
// Builtin signature (from compiler diagnostic): 
//   void __builtin_amdgcn_global_load_async_to_lds_b128(
//        v4i addrspace(1)* src, v4i addrspace(3)* dst, imm int offset, imm int cpol)
// ---------------------------------------------------------------------------
#if defined(__has_builtin)
#if __has_builtin(__builtin_amdgcn_global_load_async_to_lds_b128) && \
    __has_builtin(__builtin_amdgcn_s_wait_asynccnt)
#define HAVE_ASYNC_LDS 1
#endif
#endif

#if defined(HAVE_ASYNC_LDS)
typedef v4i as1_v4i __attribute__((address_space(1)));
typedef v4i as3_v4i __attribute__((address_space(3)));
__device__ __forceinline__ void async_b128(const float* g, const float* l) {
  __builtin_amdgcn_global_load_async_to_lds_b128(
      (as1_v4i*)(unsigned long long)g,
      (as3_v4i*)(unsigned)(unsigned long long)l,
      0, 0);
}
#endif

// Stage `n` floats (n multiple of 4*nthreads-friendly) from global to LDS.
__device__ __forceinline__ void stage_to_lds(const float* __restrict__ g,
                                             float* __restrict__ l, int n, int t,
                                             int nthreads) {
#if defined(HAVE_ASYNC_LDS)
  for (int i4 = t * 4; i4 < n; i4 += nthreads * 4) async_b128(g + i4, l + i4);
#else
  for (int i = t; i < n; i += nthreads) l[i] = g[i];
#endif
}
__device__ __forceinline__ void stage_wait() {
#if defined(HAVE_ASYNC_LDS)
  __builtin_amdgcn_s_wait_asynccnt(0);
#endif
}

// ---------------------------------------------------------------------------
// WMMA f32 16x16x4 helpers (CDNA5 VGPR layouts per ISA 7.12.2):
//  A 16x4 (MxK): lane L -> m = L%16; VGPR0 = (m, k0 + (L<16?0:2)), VGPR1 = +1
//  B 4x16 (KxN): lane L -> n = L%16; VGPR0 = (k0 + (L<16?0:2), n), VGPR1 = +1
//  C/D 16x16:    lane L -> n = L%16; VGPRr = (M = r + (L<16?0:8), n)
// ---------------------------------------------------------------------------
__device__ __forceinline__ v2f lda_f(const float* A, int ld, int r0, int k0, int lane) {
  int m  = lane & 15;
  int kk = (lane & 16) ? 2 : 0;
  v2f a;
  a.x = A[(r0 + m) * ld + k0 + kk];
  a.y = A[(r0 + m) * ld + k0 + kk + 1];
  return a;
}
__device__ __forceinline__ v2f ldb_f(const float* Bm, int ld, int k0, int c0, int lane) {
  int n  = lane & 15;
  int kk = (lane & 16) ? 2 : 0;
  v2f b;
  b.x = Bm[(k0 + kk) * ld + c0 + n];
  b.y = Bm[(k0 + kk + 1) * ld + c0 + n];
  return b;
}
// A-fragment of S^T where S is stored row-major with leading dim ld: A(m,k) = S[k][m]
__device__ __forceinline__ v2f lda_t(const float* S, int ld, int r0, int k0, int lane) {
  int m  = lane & 15;
  int kk = (lane & 16) ? 2 : 0;
  v2f a;
  a.x = S[(k0 + kk) * ld + r0 + m];
  a.y = S[(k0 + kk + 1) * ld + r0 + m];
  return a;
}
// B-fragment of S^T: B(k,n) = S[n][k]
__device__ __forceinline__ v2f ldb_t(const float* S, int ld, int k0, int c0, int lane) {
  int n  = lane & 15;
  int kk = (lane & 16) ? 2 : 0;
  v2f b;
  b.x = S[(c0 + n) * ld + k0 + kk];
  b.y = S[(c0 + n) * ld + k0 + kk + 1];
  return b;
}
__device__ __forceinline__ v8f wmma4(v2f a, v2f b, v8f c) {
  return __builtin_amdgcn_wmma_f32_16x16x4_f32(false, a, false, b, (short)0, c, false, false);
}
__device__ __forceinline__ void stc_f(float* D, int ld, int r0, int c0, int lane, v8f c) {
  int n  = lane & 15;
  int mb = (lane & 16) ? 8 : 0;
#pragma unroll
  for (int r = 0; r < 8; ++r) D[(r0 + mb + r) * ld + c0 + n] = c[r];
}

// ---------------------------------------------------------------------------
// Parallel (round-robin ordering) cyclic Jacobi for 64x64 symmetric A in LDS.
// 32 disjoint pivot pairs rotate concurrently per round, 63 rounds = 1 sweep.
// V (optional, pre-initialized to I) accumulates eigenvectors.
// ---------------------------------------------------------------------------
__device__ void jacobi64(float* A, float* V, int sweeps, int t, int nthreads,
                         int* top, int* bot, float* cs, float* sn) {
  const int NP = 32;
  if (t == 0) {
    for (int k = 0; k < NP; ++k) { top[k] = 2 * k; bot[k] = 2 * k + 1; }
  }
  __syncthreads();
  for (int sw = 0; sw < sweeps; ++sw) {
    for (int rnd = 0; rnd < 63; ++rnd) {
      if (t < NP) {
        int p = top[t], q = bot[t];
        if (p > q) { int tmp = p; p = q; q = tmp; }
        float apq = A[p * NI + q], app = A[p * NI + p], aqq = A[q * NI + q];
        float c = 1.f, s = 0.f;
        if (fabsf(apq) > 1e-12f) {
          float tau = (aqq - app) / (2.f * apq);
          float rt  = sqrtf(1.f + tau * tau);
          float tt  = (tau >= 0.f) ? 1.f / (tau + rt) : 1.f / (tau - rt);
          c = rsqrtf(1.f + tt * tt);
          s = tt * c;
        }
        cs[t] = c; sn[t] = s;
      }
      __syncthreads();
      // column phase: A <- A * G   (and V <- V * G)
      for (int item = t; item < NP * NI; item += nthreads) {
        int g = item >> 6, i = item & 63;
        int p = top[g], q = bot[g];
        if (p > q) { int tmp = p; p = q; q = tmp; }
        float c = cs[g], s = sn[g];
        float aip = A[i * NI + p], aiq = A[i * NI + q];
        A[i * NI + p] = c * aip - s * aiq;
        A[i * NI + q] = s * aip + c * aiq;
        if (V) {
          float vip = V[i * NI + p], viq = V[i * NI + q];
          V[i * NI + p] = c * vip - s * viq;
          V[i * NI + q] = s * vip + c * viq;
        }
      }
      __syncthreads();
      // row phase: A <- G^T * A
      for (int item = t; item < NP * NI; item += nthreads) {
        int g = item >> 6, i = item & 63;
        int p = top[g], q = bot[g];
        if (p > q) { int tmp = p; p = q; q = tmp; }
        float c = cs[g], s = sn[g];
        float api = A[p * NI + i], aqi = A[q * NI + i];
        A[p * NI + i] = c * api - s * aqi;
        A[q * NI + i] = s * api + c * aqi;
      }
      __syncthreads();
      // rotate round-robin tournament (top[0] fixed)
      if (t == 0) {
        int tl = top[NP - 1];
        for (int k = NP - 1; k >= 2; --k) top[k] = top[k - 1];
        int b0 = bot[0];
        for (int k = 0; k < NP - 1; ++k) bot[k] = bot[k + 1];
        bot[NP - 1] = tl;
        top[1] = b0;
      }
      __syncthreads();
    }
  }
}

// ---------------------------------------------------------------------------
// K1: sign-corrected QR == Modified Gram-Schmidt (positive diag R).
// One wave per matrix (M=4).
// ---------------------------------------------------------------------------
__global__ void qr_mgs_kernel(const float* __restrict__ W, float* __restrict__ Wo) {
  __shared__ __align__(16) float Wl[NI * NO];
  const int m = blockIdx.x;
  const int lane = threadIdx.x;   // 32 threads; each owns rows lane and lane+32
  for (int idx = lane; idx < NI * NO; idx += 32) Wl[idx] = W[m * NI * NO + idx];
  __syncthreads();
  const int r0 = lane, r1 = lane + 32;
  for (int k = 0; k < NO; ++k) {
    float v0 = Wl[r0 * NO + k], v1 = Wl[r1 * NO + k];
    float ss = v0 * v0 + v1 * v1;
    for (int off = 16; off > 0; off >>= 1) ss += __shfl_down(ss, off, 32);
    ss = __shfl(ss, 0, 32);
    float inv = rsqrtf(fmaxf(ss, 1e-30f));
    v0 *= inv; v1 *= inv;
    Wl[r0 * NO + k] = v0; Wl[r1 * NO + k] = v1;
    for (int j = k + 1; j < NO; ++j) {
      float d = v0 * Wl[r0 * NO + j] + v1 * Wl[r1 * NO + j];
      for (int off = 16; off > 0; off >>= 1) d += __shfl_down(d, off, 32);
      d = __shfl(d, 0, 32);
      Wl[r0 * NO + j] -= d * v0;
      Wl[r1 * NO + j] -= d * v1;
    }
  }
  __syncthreads();
  for (int idx = lane; idx < NI * NO; idx += 32) Wo[m * NI * NO + idx] = Wl[idx];
}

// ---------------------------------------------------------------------------
// K2: eigh(X[b,c]) via parallel Jacobi; emit logm(X) and X^{-1/2}.
// ---------------------------------------------------------------------------
__global__ void eigh_kernel(const float* __restrict__ X,
                            float* __restrict__ logX, float* __restrict__ isq) {
  __shared__ __align__(16) float A[NI * NI];
  __shared__ __align__(16) float V[NI * NI];
  __shared__ float lw[NI], iw[NI];
  __shared__ int top[32], bot[32];
  __shared__ float cs[32], sn[32];
  const int bc = blockIdx.x, t = threadIdx.x;   // 256 threads
  const float* Xg = X + (size_t)bc * NI * NI;
  stage_to_lds(Xg, A, NI * NI, t, 256);
  for (int idx = t; idx < NI * NI; idx += 256)
    V[idx] = ((idx >> 6) == (idx & 63)) ? 1.f : 0.f;
  stage_wait();
  __syncthreads();
  jacobi64(A, V, 8, t, 256, top, bot, cs, sn);
  if (t < NI) {
    float w = fmaxf(A[t * NI + t], 1e-20f);
    lw[t] = logf(w);
    iw[t] = rsqrtf(w);
  }
  __syncthreads();
  float* lo = logX + (size_t)bc * NI * NI;
  float* io = isq  + (size_t)bc * NI * NI;
  for (int idx = t; idx < NI * NI; idx += 256) {
    int x = idx >> 6, y = idx & 63;
    float a1 = 0.f, a2 = 0.f;
    for (int k = 0; k < NI; ++k) {
      float pr = V[x * NI + k] * V[y * NI + k];
      a1 += pr * lw[k];
      a2 += pr * iw[k];
    }
    lo[idx] = a1;
    io[idx] = a2;
  }
}

// ---------------------------------------------------------------------------
// K3: dis[b,i,j] = || logm( isq_i X_j isq_i ) ||_F = sqrt(sum log(lambda)^2)
// Two 64^3 GEMMs in WMMA f32, then eigenvalue-only parallel Jacobi.
// ---------------------------------------------------------------------------
__global__ void dis_kernel(const float* __restrict__ X, const float* __restrict__ isq,
                           float* __restrict__ dis) {
  __shared__ __align__(16) float Mi[NI * NI];
  __shared__ __align__(16) float Xj[NI * NI];
  __shared__ __align__(16) float U[NI * NI];
  __shared__ __align__(16) float T[NI * NI];
  __shared__ float red[NI];
  __shared__ int top[32], bot[32];
  __shared__ float cs[32], sn[32];
  const int id = blockIdx.x;                 // b*256 + i*16 + j
  const int j = id & 15, i = (id >> 4) & 15, b = id >> 8;
  const int t = threadIdx.x;                 // 256 threads, 8 waves
  if (i == j) { if (t == 0) dis[id] = 0.f; return; }
  const float* Mg = isq + (size_t)(b * C_ + i) * NI * NI;
  const float* Xg = X   + (size_t)(b * C_ + j) * NI * NI;
  stage_to_lds(Mg, Mi, NI * NI, t, 256);
  stage_to_lds(Xg, Xj, NI * NI, t, 256);
  stage_wait();
  __syncthreads();
  const int lane = t & 31, wv = t >> 5;
  // U = Mi @ Xj (16 tiles, 2 per wave)
  for (int tt = 2 * wv; tt < 2 * wv + 2; ++tt) {
    int r0 = (tt >> 2) * 16, c0 = (tt & 3) * 16;
    v8f acc = {};
    for (int k = 0; k < NI; k += 4)
      acc = wmma4(lda_f(Mi, NI, r0, k, lane), ldb_f(Xj, NI, k, c0, lane), acc);
    stc_f(U, NI, r0, c0, lane, acc);
  }
  __syncthreads();
  // T = U @ Mi
  for (int tt = 2 * wv; tt < 2 * wv + 2; ++tt) {
    int r0 = (tt >> 2) * 16, c0 = (tt & 3) * 16;
    v8f acc = {};
    for (int k = 0; k < NI; k += 4)
      acc = wmma4(lda_f(U, NI, r0, k, lane), ldb_f(Mi, NI, k, c0, lane), acc);
    stc_f(T, NI, r0, c0, lane, acc);
  }
  __syncthreads();
  jacobi64(T, (float*)0, 6, t, 256, top, bot, cs, sn);
  if (t < NI) {
    float l = logf(fmaxf(T[t * NI + t], 1e-30f));
    red[t] = l * l;
  }
  __syncthreads();
  if (t == 0) {
    float ssum = 0.f;
    for (int k = 0; k < NI; ++k) ssum += red[k];
    dis[id] = sqrtf(ssum);
  }
}

// ---------------------------------------------------------------------------
// K4: P[b, c*M+m] = Wo_m^T X[b,c] Wo_m  (WMMA f32 16x16x4 via LDS tiles)
// ---------------------------------------------------------------------------
__global__ void p_kernel(const float* __restrict__ X, const float* __restrict__ Wo,
                         float* __restrict__ out) {
  __shared__ __align__(16) float Xl[NI * NI];
  __shared__ __align__(16) float Wl[NI * NO];
  __shared__ __align__(16) float Tl[NI * NO];
  const int bc = blockIdx.x, t = threadIdx.x;   // 256 threads
  const int b = bc >> 4, c = bc & 15;
  const float* Xg = X + (size_t)bc * NI * NI;
  stage_to_lds(Xg, Xl, NI * NI, t, 256);
  const int lane = t & 31, wv = t >> 5;
  for (int m = 0; m < M_; ++m) {
    __syncthreads();
    stage_to_lds(Wo + m * NI * NO, Wl, NI * NO, t, 256);
    stage_wait();
    __syncthreads();
    // T = X @ Wo (64x32 -> 8 tiles, one per wave)
    {
      int r0 = (wv >> 1) * 16, c0 = (wv & 1) * 16;
      v8f acc = {};
      for (int k = 0; k < NI; k += 4)
        acc = wmma4(lda_f(Xl, NI, r0, k, lane), ldb_f(Wl, NO, k, c0, lane), acc);
      stc_f(Tl, NO, r0, c0, lane, acc);
    }
    __syncthreads();
    // P = Wo^T @ T (32x32 -> 4 tiles, waves 0..3)
    if (wv < 4) {
      int r0 = (wv >> 1) * 16, c0 = (wv & 1) * 16;
      v8f acc = {};
      for (int k = 0; k < NI; k += 4)
        acc = wmma4(lda_t(Wl, NO, r0, k, lane), ldb_f(Tl, NO, k, c0, lane), acc);
      float* Pg = out + (size_t)(b * (C_ * M_) + c * M_ + m) * NO * NO;
      stc_f(Pg, NO, r0, c0, lane, acc);
    }
  }
}

// ---------------------------------------------------------------------------
// K5: L[b,m] = sum_{i!=j} dis[b,i,j] * (D Wo_m)(D Wo_m)^T,  D = logX_i - logX_j.
// Fold dis via sqrt scaling of E = D@Wo; keep L accumulators in WMMA VGPRs
// across the whole (i,j) loop.
// ---------------------------------------------------------------------------
__global__ void l_kernel(const float* __restrict__ logX, const float* __restrict__ Wo,
                         const float* __restrict__ dis, float* __restrict__ out) {
  __shared__ __align__(16) float Wl[NI * NO];
  __shared__ __align__(16) float D[NI * NI];
  __shared__ __align__(16) float E[NI * NO];
  const int id = blockIdx.x;                 // b*M + m
  const int b = id >> 2, m = id & 3;
  const int t = threadIdx.x, lane = t & 31, wv = t >> 5;
  stage_to_lds(Wo + m * NI * NO, Wl, NI * NO, t, 256);
  stage_wait();
  // wave wv owns L tiles 2wv and 2wv+1 (64x64 = 16 tiles of 16x16)
  const int t0 = 2 * wv, t1 = 2 * wv + 1;
  const int r0a = (t0 >> 2) * 16, c0a = (t0 & 3) * 16;
  const int r0b = (t1 >> 2) * 16, c0b = (t1 & 3) * 16;
  v8f acc0 = {}, acc1 = {};
  for (int i = 0; i < C_; ++i) {
    for (int j = 0; j < C_; ++j) {
      if (i == j) continue;                  // D==0, dis==0: contributes nothing
      float s = sqrtf(fmaxf(dis[(b * C_ + i) * C_ + j], 0.f));
      const float* Li = logX + (size_t)(b * C_ + i) * NI * NI;
      const float* Lj = logX + (size_t)(b * C_ + j) * NI * NI;
      __syncthreads();                       // prior E reads done before rewrites
      for (int idx = t; idx < NI * NI; idx += 256) D[idx] = Li[idx] - Lj[idx];
      __syncthreads();
      // E = sqrt(dis) * (D @ Wo): 8 tiles, one per wave
      {
        int r0 = (wv >> 1) * 16, c0 = (wv & 1) * 16;
        v8f acc = {};
        for (int k = 0; k < NI; k += 4)
          acc = wmma4(lda_f(D, NI, r0, k, lane), ldb_f(Wl, NO, k, c0, lane), acc);
#pragma unroll
        for (int r = 0; r < 8; ++r) acc[r] *= s;
        stc_f(E, NO, r0, c0, lane, acc);
      }
      __syncthreads();
      // L_tiles += E @ E^T  (K = 32)
      for (int k = 0; k < NO; k += 4) {
        acc0 = wmma4(lda_f(E, NO, r0a, k, lane), ldb_t(E, NO, k, c0a, lane), acc0);
        acc1 = wmma4(lda_f(E, NO, r0b, k, lane), ldb_t(E, NO, k, c0b, lane), acc1);
      }
    }
  }
  float* Lg = out + (size_t)(B_ * C_ * M_ * NO * NO) + (size_t)(b * M_ + m) * NI * NI;
  stc_f(Lg, NI, r0a, c0a, lane, acc0);
  stc_f(Lg, NI, r0b, c0b, lane, acc1);
}

// ---------------------------------------------------------------------------
extern "C" void kernel_launch(void* const* d_in, const int* in_sizes, int n_in,
                              void* d_out, int out_size, void* d_ws, size_t ws_size,
                              hipStream_t stream) {
  (void)in_sizes; (void)n_in; (void)out_size; (void)ws_size;
  const float* X = (const float*)d_in[0];   // (B, C, NI, NI)
  const float* W = (const float*)d_in[1];   // (M, NI, NO)
  float* out = (float*)d_out;               // P (B,C*M,NO,NO) ++ L (B,M,NI,NI)
  float* ws = (float*)d_ws;
  float* Wo   = ws;                                         // M*NI*NO       =    8192 f
  float* logX = Wo + (size_t)M_ * NI * NO;                  // B*C*NI*NI     = 2097152 f
  float* isq  = logX + (size_t)B_ * C_ * NI * NI;           // B*C*NI*NI     = 2097152 f
  float* dis  = isq  + (size_t)B_ * C_ * NI * NI;           // B*C*C         =    8192 f

  qr_mgs_kernel<<<M_,            32, 0, stream>>>(W, Wo);
  eigh_kernel  <<<B_ * C_,      256, 0, stream>>>(X, logX, isq);
  dis_kernel   <<<B_ * C_ * C_, 256, 0, stream>>>(X, isq, dis);
  p_kernel     <<<B_ * C_,      256, 0, stream>>>(X, Wo, out);
  l_kernel     <<<B_ * M_,      256, 0, stream>>>(logX, Wo, dis, out);
}